// Correlation_84086869721443
// MI455X (gfx1250) — compile-verified
//
#include <hip/hip_runtime.h>

// Correlation (FlowNet-style) on gfx1250 via V_WMMA_F32_16X16X4_F32.
//
// out[b, d=(iy,ix), h, w] = (1/256) * sum_c in1[b,c,h,w] * in2[b,c,h+dy,w+dx]
//   dy = -20 + 2*iy, dx = -20 + 2*ix, iy,ix in [0,21), zero-padded OOB.
//
// Per (b, h, dy, 16-pixel strip w0..w0+15): compute the cross-matrix
// G[p,q] = sum_c in1[c, w0+p] * in2[c, h+dy, q] over the 64-column window
// q in [w0-24, w0+39] as a 16x64x256 fp32 GEMM (four 16x16 WMMA tiles).
// Every displacement is a diagonal-band element: out[p, dx] = G[p, p+dx+24].
// The 1/256 normalization and the zero-padding mask are folded into a
// loop-invariant per-column scale applied to B.

typedef float v2f __attribute__((ext_vector_type(2)));
typedef float v8f __attribute__((ext_vector_type(8)));

#define BN 16
#define CN 256
#define HN 48
#define WN 64
#define HWN (HN * WN)      // 3072 floats per channel plane
#define ND 21              // displacements per axis
#define DN (ND * ND)       // 441

__global__ __launch_bounds__(256) void corr_wmma_f32_kernel(
    const float* __restrict__ in1,
    const float* __restrict__ in2,
    float* __restrict__ out)
{
    const int wave = threadIdx.x >> 5;
    const int lane = threadIdx.x & 31;
    const int half = lane >> 4;   // 0: lanes 0-15, 1: lanes 16-31
    const int l    = lane & 15;

    // Task decode: iy fastest so the 8 waves of a block share (b,h,strip)
    // and therefore the in1 strip addresses (L0 reuse).
    int task = blockIdx.x * (blockDim.x >> 5) + wave;
    const int iy    = task % ND; task /= ND;
    const int strip = task % 4;  task /= 4;
    const int h     = task % HN;
    const int b     = task / HN;

    const int w0    = strip * 16;
    const int qbase = w0 - 24;            // 64-col window [w0-24, w0+39]
    const int dy    = -20 + 2 * iy;
    const int h2    = h + dy;
    const bool row_valid = (h2 >= 0) && (h2 < HN);   // wave-uniform

    v8f acc0 = {}, acc1 = {}, acc2 = {}, acc3 = {};

    if (row_valid) {
        // fp32 16x4 A layout: lanes 0-15 hold K=0,1 ; lanes 16-31 hold K=2,3.
        const int k0 = half * 2;

        // A fragment pointer: in1[b, k0, h, w0+l]; second K at +HWN.
        const float* __restrict__ ap =
            in1 + (((size_t)b * CN + k0) * HN + h) * WN + (w0 + l);

        // B fragment pointers per 16-col tile (mirror layout: lane = N,
        // halves split K). Scalar pointer variables (not an array) so the
        // address-space inference keeps them in the global path.
        // Invalid columns: clamp address, zero via the per-column scale.
        // Valid columns: scale = 1/256 (folds the output normalization).
        const size_t brow = (((size_t)b * CN + k0) * HN + h2) * WN;

        const int q0 = qbase + l;
        const int q1 = q0 + 16;
        const int q2 = q0 + 32;
        const int q3 = q0 + 48;
        const bool v0 = (q0 >= 0) && (q0 < WN);
        const bool v1 = (q1 >= 0) && (q1 < WN);
        const bool v2 = (q2 >= 0) && (q2 < WN);
        const bool v3 = (q3 >= 0) && (q3 < WN);
        const float inv = 1.0f / (float)CN;
        const float s0 = v0 ? inv : 0.0f;
        const float s1 = v1 ? inv : 0.0f;
        const float s2 = v2 ? inv : 0.0f;
        const float s3 = v3 ? inv : 0.0f;

        const float* __restrict__ b0p = in2 + brow + (v0 ? q0 : 0);
        const float* __restrict__ b1p = in2 + brow + (v1 ? q1 : 0);
        const float* __restrict__ b2p = in2 + brow + (v2 ? q2 : 0);
        const float* __restrict__ b3p = in2 + brow + (v3 ? q3 : 0);

#pragma unroll 4
        for (int kk = 0; kk < CN; kk += 4) {
            v2f Av = { ap[0], ap[HWN] };

            v2f B0 = { b0p[0] * s0, b0p[HWN] * s0 };
            v2f B1 = { b1p[0] * s1, b1p[HWN] * s1 };
            v2f B2 = { b2p[0] * s2, b2p[HWN] * s2 };
            v2f B3 = { b3p[0] * s3, b3p[HWN] * s3 };

            // 8 args: (neg_a, A, neg_b, B, c_mod, C, reuse_a, reuse_b)
            acc0 = __builtin_amdgcn_wmma_f32_16x16x4_f32(
                       false, Av, false, B0, (short)0, acc0, false, false);
            acc1 = __builtin_amdgcn_wmma_f32_16x16x4_f32(
                       false, Av, false, B1, (short)0, acc1, false, false);
            acc2 = __builtin_amdgcn_wmma_f32_16x16x4_f32(
                       false, Av, false, B2, (short)0, acc2, false, false);
            acc3 = __builtin_amdgcn_wmma_f32_16x16x4_f32(
                       false, Av, false, B3, (short)0, acc3, false, false);

            ap  += 4 * HWN;
            b0p += 4 * HWN;
            b1p += 4 * HWN;
            b2p += 4 * HWN;
            b3p += 4 * HWN;
        }
    }
    // else: accumulators stay zero; still store (output must be fully written).

    // Extract diagonal band. C/D layout: VGPR r, lanes 0-15 -> (M=r,   N=l),
    //                                            lanes 16-31 -> (M=r+8, N=l).
    // Scale already folded into B; stores are exact.
#pragma unroll
    for (int n = 0; n < 4; ++n) {
        const v8f acc = (n == 0) ? acc0 : (n == 1) ? acc1
                       : (n == 2) ? acc2 : acc3;
#pragma unroll
        for (int r = 0; r < 8; ++r) {
            const int m    = r + 8 * half;      // pixel p = w0 + m
            const int ncol = 16 * n + l;        // window column
            const int dx   = ncol - m - 24;     // displacement
            if (dx >= -20 && dx <= 20 && ((dx & 1) == 0)) {
                const int ix = (dx + 20) >> 1;
                const int d  = iy * ND + ix;
                const size_t idx =
                    (((size_t)b * DN + d) * HN + h) * WN + (size_t)(w0 + m);
                out[idx] = acc[r];
            }
        }
    }
}

extern "C" void kernel_launch(void* const* d_in, const int* in_sizes, int n_in,
                              void* d_out, int out_size, void* d_ws, size_t ws_size,
                              hipStream_t stream) {
    (void)in_sizes; (void)n_in; (void)out_size; (void)d_ws; (void)ws_size;
    const float* in1 = (const float*)d_in[0];
    const float* in2 = (const float*)d_in[1];
    float* out = (float*)d_out;

    // tasks = B * H * (W/16) * ND = 16*48*4*21 = 64512; 8 waves per block.
    const int tasks = BN * HN * (WN / 16) * ND;
    const int waves_per_block = 8;                 // 256 threads, wave32
    const int blocks = tasks / waves_per_block;    // 8064, exact
    corr_wmma_f32_kernel<<<blocks, waves_per_block * 32, 0, stream>>>(in1, in2, out);
}